// AdditiveMSSDLoss_25451976196369
// MI455X (gfx1250) — compile-verified
//
#include <hip/hip_runtime.h>

typedef float v2f __attribute__((ext_vector_type(2)));
typedef float v4f __attribute__((ext_vector_type(4)));
typedef float v8f __attribute__((ext_vector_type(8)));

#define NBLK 4096
#define NTHR 256

__device__ __forceinline__ void quat_rows(v4f q, float r[9]) {
    // q = (w, x, y, z)
    float w = q.x, x = q.y, y = q.z, z = q.w;
    float inv = rsqrtf(w * w + x * x + y * y + z * z);
    w *= inv; x *= inv; y *= inv; z *= inv;
    float xx = x * x, yy = y * y, zz = z * z;
    float wx = w * x, wy = w * y, wz = w * z;
    float xy = x * y, xz = x * z, yz = y * z;
    r[0] = 1.f - 2.f * (yy + zz); r[1] = 2.f * (xy - wz);       r[2] = 2.f * (xz + wy);
    r[3] = 2.f * (xy + wz);       r[4] = 1.f - 2.f * (xx + zz); r[5] = 2.f * (yz - wx);
    r[6] = 2.f * (xz - wy);       r[7] = 2.f * (yz + wx);       r[8] = 1.f - 2.f * (xx + yy);
}

// Stage 1: bandwidth-bound streaming reduce. 64 B/sample, non-temporal loads,
// deterministic hierarchical reduction (no float atomics).
__global__ __launch_bounds__(NTHR) void mssd_stage1(
    const float* __restrict__ pos_p, const float* __restrict__ rot_p,
    const float* __restrict__ pos_t, const float* __restrict__ rot_t,
    const float* __restrict__ max_r, const float* __restrict__ diam,
    float* __restrict__ part, int n) {

    const v4f* qp = (const v4f*)rot_p;
    const v4f* qt = (const v4f*)rot_t;

    float at = 0.f, ap = 0.f, ar = 0.f;
    for (int i = blockIdx.x * blockDim.x + threadIdx.x; i < n;
         i += gridDim.x * blockDim.x) {
        // positions: [B,3] f32 -> 3 coalesced dword NT streams
        float dx = __builtin_nontemporal_load(pos_p + 3 * i + 0) -
                   __builtin_nontemporal_load(pos_t + 3 * i + 0);
        float dy = __builtin_nontemporal_load(pos_p + 3 * i + 1) -
                   __builtin_nontemporal_load(pos_t + 3 * i + 1);
        float dz = __builtin_nontemporal_load(pos_p + 3 * i + 2) -
                   __builtin_nontemporal_load(pos_t + 3 * i + 2);
        float pos_err = sqrtf(dx * dx + dy * dy + dz * dz);

        // quaternions: [B,4] f32 -> b128 NT loads
        v4f q1 = __builtin_nontemporal_load(qp + i);
        v4f q2 = __builtin_nontemporal_load(qt + i);
        float Rp[9], Rt[9];
        quat_rows(q1, Rp);
        quat_rows(q2, Rt);
        float trace = 0.f;
#pragma unroll
        for (int k = 0; k < 9; ++k) trace = fmaf(Rp[k], Rt[k], trace);

        float ct = fminf(1.f, fmaxf(-1.f, 0.5f * (trace - 1.f)));
        // sin(arccos(ct)/2) == sqrt((1-ct)/2), exact for theta in [0,pi]
        float sh = sqrtf(0.5f * (1.f - ct));
        float mr = __builtin_nontemporal_load(max_r + i);
        float rot_err = 2.f * mr * sh;

        float invd = 1.f / __builtin_nontemporal_load(diam + i);
        pos_err *= invd;
        rot_err *= invd;

        ap += pos_err;
        ar += rot_err;
        at += pos_err + rot_err;
    }

    // wave32 butterfly reduce
#pragma unroll
    for (int off = 16; off; off >>= 1) {
        at += __shfl_xor(at, off);
        ap += __shfl_xor(ap, off);
        ar += __shfl_xor(ar, off);
    }

    __shared__ float sm[3][NTHR / 32];
    int lane = threadIdx.x & 31;
    int wid  = threadIdx.x >> 5;
    if (lane == 0) { sm[0][wid] = at; sm[1][wid] = ap; sm[2][wid] = ar; }
    __syncthreads();

    if (wid == 0) {
        const int nw = NTHR / 32;  // 8
        float t2 = (lane < nw) ? sm[0][lane] : 0.f;
        float p2 = (lane < nw) ? sm[1][lane] : 0.f;
        float r2 = (lane < nw) ? sm[2][lane] : 0.f;
#pragma unroll
        for (int off = nw >> 1; off; off >>= 1) {
            t2 += __shfl_xor(t2, off);
            p2 += __shfl_xor(p2, off);
            r2 += __shfl_xor(r2, off);
        }
        if (lane == 0) {
            part[0 * NBLK + blockIdx.x] = t2;
            part[1 * NBLK + blockIdx.x] = p2;
            part[2 * NBLK + blockIdx.x] = r2;
        }
    }
}

// Stage 2: one wave per output scalar. Per-lane float2 accumulation of the
// 4096 block partials, then a single full-precision f32 WMMA
// (V_WMMA_F32_16X16X4_F32, B = ones) collapses the 16x4 A tile held by the
// wave into row sums; finish with an 8-VGPR add + one cross-half shuffle.
__global__ __launch_bounds__(96) void mssd_stage2(
    const float* __restrict__ part, float* __restrict__ out, float inv_n) {
    int lane = threadIdx.x & 31;
    int w    = threadIdx.x >> 5;  // 0: total, 1: pos, 2: rot

    const v2f* src = (const v2f*)(part + w * NBLK);
    v2f acc = {0.f, 0.f};
#pragma unroll 4
    for (int c = 0; c < NBLK / 64; ++c)
        acc += src[c * 32 + lane];

    v2f ones = {1.f, 1.f};
    v8f cz = {};
    // D[m][n] = sum_k A[m][k] * 1 + 0  -> each D row holds its A-row sum
    v8f d = __builtin_amdgcn_wmma_f32_16x16x4_f32(
        /*neg_a=*/false, acc, /*neg_b=*/false, ones,
        /*c_mod=*/(short)0, cz, /*reuse_a=*/false, /*reuse_b=*/false);

    float s = d[0] + d[1] + d[2] + d[3] + d[4] + d[5] + d[6] + d[7];
    s += __shfl_xor(s, 16);  // combine rows 0-7 half with rows 8-15 half
    if (lane == 0) out[w] = s * inv_n;
}

extern "C" void kernel_launch(void* const* d_in, const int* in_sizes, int n_in,
                              void* d_out, int out_size, void* d_ws, size_t ws_size,
                              hipStream_t stream) {
    const float* pos_p = (const float*)d_in[0];
    const float* rot_p = (const float*)d_in[1];
    const float* pos_t = (const float*)d_in[2];
    const float* rot_t = (const float*)d_in[3];
    const float* max_r = (const float*)d_in[4];
    const float* diam  = (const float*)d_in[5];
    int n = in_sizes[4];  // B (max_radius element count)

    float* part = (float*)d_ws;  // 3 * NBLK floats = 48 KB scratch

    mssd_stage1<<<NBLK, NTHR, 0, stream>>>(pos_p, rot_p, pos_t, rot_t,
                                           max_r, diam, part, n);
    mssd_stage2<<<1, 96, 0, stream>>>(part, (float*)d_out, 1.0f / (float)n);
}